// GATLayer_26414048870624
// MI455X (gfx1250) — compile-verified
//
#include <hip/hip_runtime.h>
#include <hip/hip_bf16.h>
#include <math.h>

#define N     8192
#define DIN   256
#define DOUT  256
#define MAXNNZ 512   // expected ~33 nnz/row (p=0.004); 512 is >> max over 8192 rows

typedef __attribute__((ext_vector_type(2))) float v2f;
typedef __attribute__((ext_vector_type(4))) float v4f;
typedef __attribute__((ext_vector_type(8))) float v8f;

// ---------------------------------------------------------------------------
// Kernel 1: x = h @ W  (8192x256 @ 256x256, fp32) via V_WMMA_F32_16X16X4_F32.
// One wave32 per 16x16 C tile; 8 waves per block.
//
// f32 A-matrix 16x4 layout (ISA 7.12.2): lanes 0-15 -> K=0(V0),K=1(V1);
// lanes 16-31 -> K=2(V0),K=3(V1).  M = lane%16.
// f32 B-matrix 4x16: N = lane%16; lanes 0-15 hold K={0,1}, lanes 16-31 K={2,3}.
// f32 C/D 16x16: VGPR v holds M = v + 8*(lane/16), N = lane%16.
// ---------------------------------------------------------------------------
__global__ __launch_bounds__(256)
void gat_gemm_hw(const float* __restrict__ h,
                 const float* __restrict__ w,
                 float* __restrict__ x) {
    const int lane   = threadIdx.x & 31;
    const int waveId = threadIdx.x >> 5;
    const int tileId = blockIdx.x * 8 + waveId;          // 8192 tiles total
    const int tm = tileId >> 4;                          // 512 M-tiles
    const int tn = tileId & 15;                          // 16  N-tiles
    const int m0 = tm * 16;
    const int n0 = tn * 16;

    const int lm = lane & 15;      // M (for A) / N (for B,C)
    const int hi = lane >> 4;      // 0: K={0,1}, 1: K={2,3}

    v8f acc = {};

    const float* arow = h + (size_t)(m0 + lm) * DIN;     // A row for this lane

    #pragma unroll 4
    for (int k = 0; k < DIN; k += 4) {
        // A fragment: contiguous float2 from h row at K = k + 2*hi
        v2f a = *(const v2f*)(arow + k + 2 * hi);
        // B fragment: W[(k+2*hi+q)][n0+lm], q = 0,1
        v2f b;
        b.x = w[(size_t)(k + 2 * hi    ) * DOUT + n0 + lm];
        b.y = w[(size_t)(k + 2 * hi + 1) * DOUT + n0 + lm];
        // D = A*B + C   (8-arg form: neg_a, A, neg_b, B, c_mod, C, reuseA, reuseB)
        acc = __builtin_amdgcn_wmma_f32_16x16x4_f32(
                  false, a, false, b, (short)0, acc, false, false);
    }

    // Store C tile
    #pragma unroll
    for (int v = 0; v < 8; ++v) {
        x[(size_t)(m0 + v + hi * 8) * DOUT + n0 + lm] = acc[v];
    }
}

// ---------------------------------------------------------------------------
// Kernel 2: el[i] = x[i,:]·a_l, er[i] = x[i,:]·a_r.  One block per row.
// ---------------------------------------------------------------------------
__global__ __launch_bounds__(256)
void gat_logits(const float* __restrict__ x,
                const float* __restrict__ a_l,
                const float* __restrict__ a_r,
                float* __restrict__ el,
                float* __restrict__ er) {
    const int i = blockIdx.x;
    const int t = threadIdx.x;
    __shared__ float sl[256];
    __shared__ float sr[256];
    const float xv = x[(size_t)i * DOUT + t];
    sl[t] = xv * a_l[t];
    sr[t] = xv * a_r[t];
    __syncthreads();
    for (int s = 128; s > 0; s >>= 1) {
        if (t < s) { sl[t] += sl[t + s]; sr[t] += sr[t + s]; }
        __syncthreads();
    }
    if (t == 0) { el[i] = sl[0]; er[i] = sr[0]; }
}

// ---------------------------------------------------------------------------
// Kernel 3: per-row masked-softmax aggregation.
// One 256-thread block per row i.
//   * adj row scanned with NON-TEMPORAL b128 loads: 268 MB single-use stream
//     must not sweep the 192 MB L2, so that x (8 MB) stays L2-resident for
//     the per-nnz gathers.
//   * nnz (j, exp(leaky_relu(el_i+er_j))) appended into LDS, denom via
//     LDS float atomic (ds_add_f32).
//   * thread t owns channel t: out[i,t] = (sum_p w_p * x[j_p,t])/denom + b[t]
//     stored non-temporally (write-once output).
// ---------------------------------------------------------------------------
__global__ __launch_bounds__(256)
void gat_aggregate(const float* __restrict__ adj,
                   const float* __restrict__ x,
                   const float* __restrict__ el,
                   const float* __restrict__ er,
                   const float* __restrict__ bias,
                   float* __restrict__ out) {
    const int i = blockIdx.x;
    const int t = threadIdx.x;

    __shared__ int   s_idx[MAXNNZ];
    __shared__ float s_w[MAXNNZ];
    __shared__ int   s_cnt;
    __shared__ float s_denom;

    if (t == 0) { s_cnt = 0; s_denom = 0.0f; }
    __syncthreads();

    const float eli = el[i];
    const v4f* __restrict__ arow4 = (const v4f*)(adj + (size_t)i * N);

    // 8192 floats / (256 threads * 4) = 8 iterations; coalesced 4KB segments.
    for (int q = t; q < N / 4; q += 256) {
        const v4f a4 = __builtin_nontemporal_load(arow4 + q);  // TH=NT stream
        #pragma unroll
        for (int e = 0; e < 4; ++e) {
            if (a4[e] != 0.0f) {
                const int   j  = q * 4 + e;
                const float s  = eli + er[j];
                const float ls = (s > 0.0f) ? s : 0.2f * s;    // leaky_relu(0.2)
                const float wv = __expf(ls);
                const int p = atomicAdd(&s_cnt, 1);
                if (p < MAXNNZ) { s_idx[p] = j; s_w[p] = wv; }
                atomicAdd(&s_denom, wv);                       // ds_add_f32
            }
        }
    }
    __syncthreads();

    const int   cnt = (s_cnt < MAXNNZ) ? s_cnt : MAXNNZ;
    const float inv = 1.0f / fmaxf(s_denom, 1e-12f);

    float acc = 0.0f;
    for (int p = 0; p < cnt; ++p) {
        acc = fmaf(s_w[p], x[(size_t)s_idx[p] * DOUT + t], acc);  // L2-hit gathers
    }
    __builtin_nontemporal_store(acc * inv + bias[t], &out[(size_t)i * DOUT + t]);
}

// ---------------------------------------------------------------------------
extern "C" void kernel_launch(void* const* d_in, const int* in_sizes, int n_in,
                              void* d_out, int out_size, void* d_ws, size_t ws_size,
                              hipStream_t stream) {
    const float* h      = (const float*)d_in[0];   // [N, DIN]
    const float* adj    = (const float*)d_in[1];   // [N, N]
    const float* weight = (const float*)d_in[2];   // [DIN, DOUT]
    const float* a_l    = (const float*)d_in[3];   // [DOUT]
    const float* a_r    = (const float*)d_in[4];   // [DOUT]
    const float* bias   = (const float*)d_in[5];   // [DOUT]
    float*       out    = (float*)d_out;           // [N, DOUT]

    // workspace layout: x [N*DOUT] | el [N] | er [N]
    float* x  = (float*)d_ws;
    float* el = x + (size_t)N * DOUT;
    float* er = el + N;

    // 1) x = h @ W : 8192 16x16 tiles, 8 waves/block -> 1024 blocks
    gat_gemm_hw<<<dim3(1024), dim3(256), 0, stream>>>(h, weight, x);
    // 2) attention logits
    gat_logits<<<dim3(N), dim3(256), 0, stream>>>(x, a_l, a_r, el, er);
    // 3) masked softmax + sparse aggregation (streams 268 MB of adj -> ~11.5 us floor)
    gat_aggregate<<<dim3(N), dim3(256), 0, stream>>>(adj, x, el, er, bias, out);
}